// EGT_11622181503376
// MI455X (gfx1250) — compile-verified
//
#include <hip/hip_runtime.h>
#include <hip/hip_bf16.h>

// EGT attention forward for MI455X (gfx1250, wave32, WMMA + async-to-LDS).
// L=3072, H=8, D=64. Memory-bound (~1.2GB mandatory traffic); fp32 WMMA
// (V_WMMA_F32_16X16X4_F32) for QK^T and A_tild@V. Two m-segments per l-tile
// (384 blocks/kernel); tiles stream into double-buffered LDS with
// GLOBAL_LOAD_ASYNC_TO_LDS_B128 (ASYNCcnt), overlapping HBM latency with
// compute. V_att combined via 2 hardware f32 atomics per element (bitwise
// order-independent); softmax stats via per-segment partials + merge pass.

#define LSEQ  3072
#define NH    8
#define HD    64
#define C3    1536            // 3*D*H floats per QKV row
#define ER    (LSEQ*NH)       // 24576 floats per l-row of E/G/Hhat/Atild
#define LH    (LSEQ*NH)       // stats entries
#define MT    (LSEQ/16)       // 192 m-tiles total
#define NSEG  2
#define SEGMT (MT/NSEG)       // 96 m-tiles per segment
#define KSTR  516             // LDS stride, 512-float rows (bank pad, 16B align)
#define ESTR  132             // LDS stride, 128-float rows (bank pad, 16B align)
#define SZK   (16*KSTR)       // one K/V tile buffer (floats)
#define SZE   (16*ESTR)       // one E/H/G tile buffer (floats)
#define QKSCALE 0.125f        // 1/sqrt(64)

typedef __attribute__((ext_vector_type(2))) float v2f;
typedef __attribute__((ext_vector_type(4))) float f4v;
typedef __attribute__((ext_vector_type(8))) float v8f;

#define ASYNC_WAIT(imm) asm volatile("s_wait_asynccnt " imm ::: "memory")

// One 16-byte async global->LDS copy (ASYNCcnt-tracked, no VGPR staging).
__device__ __forceinline__ void async_b128(float* ldsDst, const float* gSrc) {
  asm volatile("global_load_async_to_lds_b128 %0, %1, off"
               :: "v"((unsigned)(uintptr_t)ldsDst), "v"(gSrc)
               : "memory");
}

// ---- cooperative tile movers (256 threads, 16B, fully coalesced) ----

// 16 rows x 512 floats  global -> LDS (row stride KSTR), async
__device__ __forceinline__ void async_tile_512(float* dst, const float* srcBase,
                                               int rowStride, int tid) {
#pragma unroll
  for (int j = 0; j < 8; ++j) {
    int f   = tid + j * 256;
    int row = f >> 7;
    int c4  = f & 127;
    async_b128(dst + row * KSTR + c4 * 4,
               srcBase + (size_t)row * rowStride + c4 * 4);
  }
}

// 16 rows x 128 floats  global -> LDS (row stride ESTR), async
__device__ __forceinline__ void async_tile_128(float* dst, const float* srcBase,
                                               int rowStride, int tid) {
#pragma unroll
  for (int j = 0; j < 2; ++j) {
    int f   = tid + j * 256;
    int row = f >> 5;
    int c4  = f & 31;
    async_b128(dst + row * ESTR + c4 * 4,
               srcBase + (size_t)row * rowStride + c4 * 4);
  }
}

// 16 rows x 128 floats  LDS -> global; optional nontemporal
template <bool NT>
__device__ __forceinline__ void coop_store_128(float* dstBase, int rowStride,
                                               const float* src, int tid) {
#pragma unroll
  for (int j = 0; j < 2; ++j) {
    int f   = tid + j * 256;
    int row = f >> 5;
    int c4  = f & 31;
    f4v v = *(const f4v*)(src + row * ESTR + c4 * 4);
    f4v* p = (f4v*)(dstBase + (size_t)row * rowStride + c4 * 4);
    if (NT) __builtin_nontemporal_store(v, p); else *p = v;
  }
}

// =====================================================================
// Kernel 1: H_hat = clip(Q K^T * scale) + E over one m-segment, plus
// per-(l,h) partial online softmax stats -> ws[seg].
// Block = 16 l-rows x all 8 heads x 1 m-segment; wave w == head w.
// Double-buffered LDS, async prefetch of tile t+1 during tile t compute.
// =====================================================================
__global__ __launch_bounds__(256) void egt_qk_hhat_stats(
    const float* __restrict__ QKV, const float* __restrict__ E,
    float* __restrict__ Hhat, float* __restrict__ pmax,
    float* __restrict__ psum) {
  __shared__ float sK[2 * SZK];    // K tile [m_local][d*8+h], 2 buffers
  __shared__ float sEH[2 * SZE];   // E tile -> H_hat in place, 2 buffers

  const int tid  = threadIdx.x;
  const int lane = tid & 31;
  const int h    = tid >> 5;
  const int l0   = blockIdx.x * 16;
  const int seg  = blockIdx.y;
  const int lm   = lane & 15;
  const int lh   = lane >> 4;
  const int mt0  = seg * SEGMT;

  // Q A-operands for all 16 K-chunks: M = lane&15, K = 2*(lane>>4)+{x,y}.
  v2f a[16];
  {
    const float* qb = QKV + (size_t)(l0 + lm) * C3 + lh * 16 + h;
#pragma unroll
    for (int kc = 0; kc < 16; ++kc) {
      a[kc].x = qb[kc * 32];
      a[kc].y = qb[kc * 32 + 8];
    }
  }

  float rmax[8], rsum[8];
#pragma unroll
  for (int r = 0; r < 8; ++r) { rmax[r] = -__builtin_inff(); rsum[r] = 0.0f; }

  const int sKb = lm * KSTR + lh * 16 + h;

  // Prologue: stream first tile into buffer 0.
  async_tile_512(sK, QKV + (size_t)(mt0 * 16) * C3 + 512, C3, tid);
  async_tile_128(sEH, E + (size_t)l0 * ER + (size_t)(mt0 * 16) * NH, ER, tid);

  for (int i = 0; i < SEGMT; ++i) {
    const int m0 = (mt0 + i) * 16;
    const int p  = i & 1;
    float* sKp  = sK + p * SZK;
    float* sEHp = sEH + p * SZE;

    // Order prior-iteration LDS reads before overwriting the other buffer.
    __syncthreads();
    if (i + 1 < SEGMT) {
      const int m1 = m0 + 16;
      async_tile_512(sK + (p ^ 1) * SZK, QKV + (size_t)m1 * C3 + 512, C3, tid);
      async_tile_128(sEH + (p ^ 1) * SZE,
                     E + (size_t)l0 * ER + (size_t)m1 * NH, ER, tid);
      ASYNC_WAIT("0xa");   // 10 ops of tile i+1 in flight; tile i resident
    } else {
      ASYNC_WAIT("0x0");
    }
    __syncthreads();

    // Two independent accumulator chains to halve the WMMA RAW chain.
    v8f acc0 = {}, acc1 = {};
#pragma unroll
    for (int kc = 0; kc < 8; ++kc) {
      v2f b0, b1;
      b0.x = sKp[sKb + kc * 32];
      b0.y = sKp[sKb + kc * 32 + 8];
      b1.x = sKp[sKb + (kc + 8) * 32];
      b1.y = sKp[sKb + (kc + 8) * 32 + 8];
      acc0 = __builtin_amdgcn_wmma_f32_16x16x4_f32(false, a[kc], false, b0,
                                                   (short)0, acc0, false, false);
      acc1 = __builtin_amdgcn_wmma_f32_16x16x4_f32(false, a[kc + 8], false, b1,
                                                   (short)0, acc1, false, false);
    }
    v8f acc = acc0 + acc1;

#pragma unroll
    for (int r = 0; r < 8; ++r) {
      int rowL = r + lh * 8;
      int eidx = rowL * ESTR + lm * 8 + h;
      float hh = fminf(fmaxf(acc[r] * QKSCALE, -5.0f), 5.0f) + sEHp[eidx];
      float nm = fmaxf(rmax[r], hh);
      rsum[r]  = rsum[r] * __expf(rmax[r] - nm) + __expf(hh - nm);
      rmax[r]  = nm;
      sEHp[eidx] = hh;
    }
    __syncthreads();
    coop_store_128<false>(Hhat + (size_t)l0 * ER + (size_t)m0 * NH, ER, sEHp,
                          tid);
  }

  // Merge partials across the 16 lanes of each half-wave.
#pragma unroll
  for (int mask = 1; mask < 16; mask <<= 1) {
#pragma unroll
    for (int r = 0; r < 8; ++r) {
      float om = __shfl_xor(rmax[r], mask, 32);
      float os = __shfl_xor(rsum[r], mask, 32);
      float nm = fmaxf(rmax[r], om);
      rsum[r]  = rsum[r] * __expf(rmax[r] - nm) + os * __expf(om - nm);
      rmax[r]  = nm;
    }
  }
  if (lane == 0 || lane == 16) {
#pragma unroll
    for (int r = 0; r < 8; ++r) {
      int row = l0 + r + lh * 8;
      pmax[seg * LH + row * NH + h] = rmax[r];
      psum[seg * LH + row * NH + h] = rsum[r];
    }
  }
}

// =====================================================================
// Stats merge: combine NSEG partial (max, sumexp) pairs per (l,h).
// =====================================================================
__global__ __launch_bounds__(256) void egt_merge_stats(
    const float* __restrict__ pmax, const float* __restrict__ psum,
    float* __restrict__ fmax, float* __restrict__ fsum) {
  int i = blockIdx.x * 256 + threadIdx.x;
  float m0 = pmax[i], m1 = pmax[LH + i];
  float s0 = psum[i], s1 = psum[LH + i];
  float nm = fmaxf(m0, m1);
  fmax[i] = nm;
  fsum[i] = s0 * __expf(m0 - nm) + s1 * __expf(m1 - nm);
}

// =====================================================================
// Kernel 2: A_tild = softmax(H_hat) * sigmoid(G) (materialized), and
// partial V_att = A_tild @ V over one m-segment, combined with hardware
// f32 atomics (exactly 2 addends/elem -> bitwise deterministic).
// Double-buffered LDS with async prefetch.
// =====================================================================
__global__ __launch_bounds__(256) void egt_atild_av(
    const float* __restrict__ QKV, const float* __restrict__ G,
    const float* __restrict__ Hhat, float* __restrict__ Atild,
    float* __restrict__ Vatt, const float* __restrict__ fmax,
    const float* __restrict__ fsum) {
  __shared__ float sV[2 * SZK];   // V tile [m_local][d*8+h], 2 buffers
  __shared__ float sH[2 * SZE];   // H_hat tile -> A_tild in place, 2 buffers
  __shared__ float sG[2 * SZE];   // G tile, 2 buffers

  const int tid  = threadIdx.x;
  const int lane = tid & 31;
  const int h    = tid >> 5;
  const int l0   = blockIdx.x * 16;
  const int seg  = blockIdx.y;
  const int lm   = lane & 15;
  const int lh   = lane >> 4;
  const int mt0  = seg * SEGMT;

  float mx[8], inv[8];
#pragma unroll
  for (int r = 0; r < 8; ++r) {
    int row = l0 + r + lh * 8;
    mx[r]  = fmax[row * NH + h];
    inv[r] = 1.0f / fsum[row * NH + h];
  }

  v8f vacc[4] = {{}, {}, {}, {}};   // 16 l-rows x 64 d cols partial

  // Prologue: stream first tile into buffer 0.
  async_tile_512(sV, QKV + (size_t)(mt0 * 16) * C3 + 1024, C3, tid);
  async_tile_128(sH, Hhat + (size_t)l0 * ER + (size_t)(mt0 * 16) * NH, ER, tid);
  async_tile_128(sG, G + (size_t)l0 * ER + (size_t)(mt0 * 16) * NH, ER, tid);

  for (int i = 0; i < SEGMT; ++i) {
    const int m0 = (mt0 + i) * 16;
    const int p  = i & 1;
    float* sVp = sV + p * SZK;
    float* sHp = sH + p * SZE;
    float* sGp = sG + p * SZE;

    __syncthreads();
    if (i + 1 < SEGMT) {
      const int m1 = m0 + 16;
      async_tile_512(sV + (p ^ 1) * SZK, QKV + (size_t)m1 * C3 + 1024, C3, tid);
      async_tile_128(sH + (p ^ 1) * SZE,
                     Hhat + (size_t)l0 * ER + (size_t)m1 * NH, ER, tid);
      async_tile_128(sG + (p ^ 1) * SZE,
                     G + (size_t)l0 * ER + (size_t)m1 * NH, ER, tid);
      ASYNC_WAIT("0xc");   // 12 ops of tile i+1 in flight; tile i resident
    } else {
      ASYNC_WAIT("0x0");
    }
    __syncthreads();

    // A_tild elementwise, in WMMA C-lane layout (row stats live in this lane).
#pragma unroll
    for (int r = 0; r < 8; ++r) {
      int rowL = r + lh * 8;
      int eidx = rowL * ESTR + lm * 8 + h;
      float at = __expf(sHp[eidx] - mx[r]) * inv[r];
      float g  = sGp[eidx];
      at *= 1.0f / (1.0f + __expf(-g));
      sHp[eidx] = at;
    }
    __syncthreads();

    coop_store_128<true>(Atild + (size_t)l0 * ER + (size_t)m0 * NH, ER, sHp,
                         tid);

    // V_att[l,d] += sum_m A_tild[l,m] * V[m,d]; 4 independent acc chains.
#pragma unroll
    for (int kc = 0; kc < 4; ++kc) {
      v2f av;
      av.x = sHp[lm * ESTR + (kc * 4 + lh * 2) * 8 + h];
      av.y = sHp[lm * ESTR + (kc * 4 + lh * 2 + 1) * 8 + h];
#pragma unroll
      for (int n = 0; n < 4; ++n) {
        v2f bv;
        int d = n * 16 + lm;
        bv.x = sVp[(kc * 4 + lh * 2) * KSTR + d * 8 + h];
        bv.y = sVp[(kc * 4 + lh * 2 + 1) * KSTR + d * 8 + h];
        vacc[n] = __builtin_amdgcn_wmma_f32_16x16x4_f32(
            false, av, false, bv, (short)0, vacc[n], false, false);
      }
    }
  }

  // Combine segment partials: hardware f32 atomic add onto zeroed V_att.
#pragma unroll
  for (int n = 0; n < 4; ++n) {
#pragma unroll
    for (int r = 0; r < 8; ++r) {
      int row = l0 + r + lh * 8;
      int d   = n * 16 + lm;
      unsafeAtomicAdd(&Vatt[(size_t)row * (HD * NH) + d * NH + h], vacc[n][r]);
    }
  }
}

extern "C" void kernel_launch(void* const* d_in, const int* in_sizes, int n_in,
                              void* d_out, int out_size, void* d_ws, size_t ws_size,
                              hipStream_t stream) {
  const float* QKV = (const float*)d_in[0];
  const float* E   = (const float*)d_in[1];
  const float* G   = (const float*)d_in[2];

  float* out   = (float*)d_out;
  float* Vatt  = out;                                   // L * 512
  float* Hhat  = out + (size_t)LSEQ * (HD * NH);        // L*L*H
  float* Atild = Hhat + (size_t)LSEQ * LSEQ * NH;       // L*L*H

  float* pmax = (float*)d_ws;                           // NSEG * LH
  float* psum = pmax + (size_t)NSEG * LH;               // NSEG * LH
  float* fmax = psum + (size_t)NSEG * LH;               // LH
  float* fsum = fmax + (size_t)LH;                      // LH

  hipMemsetAsync(Vatt, 0, (size_t)LSEQ * HD * NH * sizeof(float), stream);

  dim3 block(256);
  egt_qk_hhat_stats<<<dim3(LSEQ / 16, NSEG), block, 0, stream>>>(QKV, E, Hhat,
                                                                 pmax, psum);
  egt_merge_stats<<<dim3(LH / 256), block, 0, stream>>>(pmax, psum, fmax, fsum);
  egt_atild_av<<<dim3(LSEQ / 16, NSEG), block, 0, stream>>>(QKV, G, Hhat, Atild,
                                                            Vatt, fmax, fsum);
}